// GIN_MLP2_43293270343763
// MI455X (gfx1250) — compile-verified
//
#include <hip/hip_runtime.h>

typedef __attribute__((ext_vector_type(2))) float v2f;
typedef __attribute__((ext_vector_type(8))) float v8f;

#define N_NODES 50000
#define NROWP   50016   // padded to multiple of 32 for 2-strip WMMA waves
#define N_EDGES 800000
#define CH      256
#define BATCH   2048
#define CTXN    50
#define IDXS    53      // 3 + CTX

#define WMMA_F32(a, b, c) \
  __builtin_amdgcn_wmma_f32_16x16x4_f32(false, (a), false, (b), (short)0, (c), false, false)

__device__ __forceinline__ v2f ld2(const float* p) { return *(const v2f*)p; }

// ---------------------------------------------------------------- utilities
__global__ void zero_kernel(float* __restrict__ p, long n) {
  long i = (long)blockIdx.x * blockDim.x + threadIdx.x;
  if (i < n) p[i] = 0.0f;
}

__global__ void deg_kernel(const int* __restrict__ dst, float* __restrict__ deg) {
  int e = blockIdx.x * blockDim.x + threadIdx.x;
  if (e < N_EDGES) atomicAdd(&deg[dst[e]], 1.0f);
}

// msg[dst] += w * x[src]   (one thread = one edge x 4 channels)
__global__ void scatter_kernel(const float* __restrict__ x, const int* __restrict__ src,
                               const int* __restrict__ dst, const float* __restrict__ w,
                               float* __restrict__ msg) {
  long tid = (long)blockIdx.x * blockDim.x + threadIdx.x;
  if (tid >= (long)N_EDGES * 64) return;
  int e = (int)(tid >> 6);
  int q = (int)(tid & 63) << 2;
  int s = src[e], d = dst[e];
  float we = w[e];
  const float4 v = *(const float4*)(x + (long)s * CH + q);
  float* mp = msg + (long)d * CH + q;
  atomicAdd(mp + 0, we * v.x);
  atomicAdd(mp + 1, we * v.y);
  atomicAdd(mp + 2, we * v.z);
  atomicAdd(mp + 3, we * v.w);
}

// msg = (1+eps)*x + msg/max(deg,1)   (in place, float4 per thread; rows < N_NODES)
__global__ void h_kernel(const float* __restrict__ x, const float* __restrict__ deg,
                         const float* __restrict__ eps, int eidx, float* __restrict__ msg) {
  long tid = (long)blockIdx.x * blockDim.x + threadIdx.x;
  if (tid >= (long)N_NODES * (CH / 4)) return;
  long base = tid * 4;
  int node = (int)(tid >> 6);
  float denom = fmaxf(deg[node], 1.0f);
  float e = 1.0f + eps[eidx];
  float4 xv = *(const float4*)(x + base);
  float4 mv = *(const float4*)(msg + base);
  float4 r;
  r.x = e * xv.x + mv.x / denom;
  r.y = e * xv.y + mv.y / denom;
  r.z = e * xv.z + mv.z / denom;
  r.w = e * xv.w + mv.w / denom;
  *(float4*)(msg + base) = r;
}

// Interleave rows (2p, 2p+1) of W (KxN) -> Wt[p][n] = {W[2p][n], W[2p+1][n]}
// so a WMMA B fragment is one b64 load. t ranges over (K/2)*N.
__global__ void interleave_kernel(const float* __restrict__ W, float* __restrict__ Wt,
                                  int total, int N) {
  int t = blockIdx.x * blockDim.x + threadIdx.x;
  if (t >= total) return;
  int p = t / N, n = t % N;
  Wt[(long)p * 2 * N + 2 * n]     = W[(long)(2 * p) * N + n];
  Wt[(long)p * 2 * N + 2 * n + 1] = W[(long)(2 * p + 1) * N + n];
}

// --------------------- GIN GEMM: out = act(A@W + b), A:(NROWP,256), Wt interleaved
// 128 thr = 4 waves; each wave: 2 M-strips (32 rows) x 64 N  -> 8 WMMA per K-step
__global__ void __launch_bounds__(128)
gemm256_kernel(const float* __restrict__ A, const float* __restrict__ Wt,
               const float* __restrict__ bias, float* __restrict__ out,
               int relu, int zero0) {
  int wave = threadIdx.x >> 5;
  int lane = threadIdx.x & 31;
  int nl = lane & 15;
  int koff = (lane >> 4) * 2;
  int nbase = wave * 64;
  int m0 = blockIdx.x * 32;
  const float* a0r = A + (long)(m0 + nl) * CH;
  const float* a1r = A + (long)(m0 + 16 + nl) * CH;

  v8f c[2][4];
#pragma unroll
  for (int t = 0; t < 4; ++t) {
    float bb = bias[nbase + t * 16 + nl];
#pragma unroll
    for (int j = 0; j < 8; ++j) { c[0][t][j] = bb; c[1][t][j] = bb; }
  }

  for (int k = 0; k < CH; k += 4) {
    v2f a0 = ld2(a0r + k + koff);
    v2f a1 = ld2(a1r + k + koff);
    const float* wt = Wt + (long)((k + koff) >> 1) * (2 * CH);
    v2f b0 = ld2(wt + 2 * (nbase + nl));
    v2f b1 = ld2(wt + 2 * (nbase + 16 + nl));
    v2f b2 = ld2(wt + 2 * (nbase + 32 + nl));
    v2f b3 = ld2(wt + 2 * (nbase + 48 + nl));
    c[0][0] = WMMA_F32(a0, b0, c[0][0]); c[1][0] = WMMA_F32(a1, b0, c[1][0]);
    c[0][1] = WMMA_F32(a0, b1, c[0][1]); c[1][1] = WMMA_F32(a1, b1, c[1][1]);
    c[0][2] = WMMA_F32(a0, b2, c[0][2]); c[1][2] = WMMA_F32(a1, b2, c[1][2]);
    c[0][3] = WMMA_F32(a0, b3, c[0][3]); c[1][3] = WMMA_F32(a1, b3, c[1][3]);
  }

  int mhi = (lane >> 4) << 3;
#pragma unroll
  for (int s = 0; s < 2; ++s)
#pragma unroll
    for (int t = 0; t < 4; ++t)
#pragma unroll
      for (int j = 0; j < 8; ++j) {
        int mm = m0 + s * 16 + mhi + j;
        float v = c[s][t][j];
        if (relu) v = fmaxf(v, 0.f);
        if (zero0 && mm == 0) v = 0.f;
        out[(long)mm * CH + nbase + t * 16 + nl] = v;
      }
}

// ---------- pair = x[i0]@Wc1a + x[i1]@Wc1b (2048x256); Wa_t/Wb_t interleaved
__global__ void __launch_bounds__(128)
pair_kernel(const float* __restrict__ x, const int* __restrict__ indices,
            const float* __restrict__ Wc1t, float* __restrict__ pair) {
  int wave = threadIdx.x >> 5;
  int lane = threadIdx.x & 31;
  int mtile = blockIdx.x;  // 128 tiles over B=2048
  int nl = lane & 15;
  int koff = (lane >> 4) * 2;
  int nbase = wave * 64;
  int row = mtile * 16 + nl;
  int i0 = indices[row * IDXS + 0];
  int i1 = indices[row * IDXS + 1];
  const float* a0r = x + (long)i0 * CH;
  const float* a1r = x + (long)i1 * CH;
  const float* Wat = Wc1t;
  const float* Wbt = Wc1t + 256 * 256;

  v8f c0 = {}, c1 = {}, c2 = {}, c3 = {};
  for (int k = 0; k < CH; k += 4) {
    v2f a0 = ld2(a0r + k + koff);
    v2f a1 = ld2(a1r + k + koff);
    {
      const float* wt = Wat + (long)((k + koff) >> 1) * 512;
      v2f b;
      b = ld2(wt + 2 * (nbase + nl));      c0 = WMMA_F32(a0, b, c0);
      b = ld2(wt + 2 * (nbase + 16 + nl)); c1 = WMMA_F32(a0, b, c1);
      b = ld2(wt + 2 * (nbase + 32 + nl)); c2 = WMMA_F32(a0, b, c2);
      b = ld2(wt + 2 * (nbase + 48 + nl)); c3 = WMMA_F32(a0, b, c3);
    }
    {
      const float* wt = Wbt + (long)((k + koff) >> 1) * 512;
      v2f b;
      b = ld2(wt + 2 * (nbase + nl));      c0 = WMMA_F32(a1, b, c0);
      b = ld2(wt + 2 * (nbase + 16 + nl)); c1 = WMMA_F32(a1, b, c1);
      b = ld2(wt + 2 * (nbase + 32 + nl)); c2 = WMMA_F32(a1, b, c2);
      b = ld2(wt + 2 * (nbase + 48 + nl)); c3 = WMMA_F32(a1, b, c3);
    }
  }

  int mrow = mtile * 16 + ((lane >> 4) << 3);
#pragma unroll
  for (int j = 0; j < 8; ++j) {
    float* orow = pair + (long)(mrow + j) * CH;
    orow[nbase + nl]      = c0[j];
    orow[nbase + 16 + nl] = c1[j];
    orow[nbase + 32 + nl] = c2[j];
    orow[nbase + 48 + nl] = c3[j];
  }
}

// ------- fused context MLP: one block = 32 (b,ctx) rows, 3-layer chain in LDS.
// AsY1 staging buffer is reused: A rows for GEMM1, then Y1 output (after barrier).
__global__ void __launch_bounds__(128)
ctx_kernel(const float* __restrict__ x, const int* __restrict__ indices,
           const float* __restrict__ Wc1ct, const float* __restrict__ bc1,
           const float* __restrict__ pair, const float* __restrict__ Wc2t,
           const float* __restrict__ bc2, const float* __restrict__ Wc3,
           const float* __restrict__ bc3, float* __restrict__ yscal) {
  __shared__ float AsY1[32 * 256];  // 32 KB
  __shared__ float Y2[32 * 128];    // 16 KB
  int rbase = blockIdx.x * 32;      // 3200 tiles over B*CTX = 102400 rows
  int wave = threadIdx.x >> 5;
  int lane = threadIdx.x & 31;
  int nl = lane & 15;
  int koff = (lane >> 4) * 2;
  int mhi = (lane >> 4) << 3;

  // stage gathered x[ctx_node] rows: 32 x 256 floats
  for (int i = threadIdx.x; i < 32 * 64; i += 128) {
    int r = i >> 6;
    int q = (i & 63) << 2;
    int grow = rbase + r;
    int b = grow / CTXN, cc = grow % CTXN;
    int node = indices[b * IDXS + 3 + cc];
    *(float4*)(AsY1 + r * 256 + q) = *(const float4*)(x + (long)node * CH + q);
  }
  __syncthreads();

  // GEMM1: (32x256)@Wc1c + pair[b] + bc1 -> relu -> Y1 (into AsY1)
  int nbase = wave * 64;
  v8f c[2][4];
#pragma unroll
  for (int s = 0; s < 2; ++s)
#pragma unroll
    for (int t = 0; t < 4; ++t)
#pragma unroll
      for (int j = 0; j < 8; ++j) {
        int mm = s * 16 + mhi + j;
        int b = (rbase + mm) / CTXN;
        int n = nbase + t * 16 + nl;
        c[s][t][j] = pair[(long)b * CH + n] + bc1[n];
      }
  for (int k = 0; k < 256; k += 4) {
    v2f a0 = ld2(AsY1 + nl * 256 + k + koff);
    v2f a1 = ld2(AsY1 + (nl + 16) * 256 + k + koff);
    const float* wt = Wc1ct + (long)((k + koff) >> 1) * 512;
    v2f b0 = ld2(wt + 2 * (nbase + nl));
    v2f b1 = ld2(wt + 2 * (nbase + 16 + nl));
    v2f b2 = ld2(wt + 2 * (nbase + 32 + nl));
    v2f b3 = ld2(wt + 2 * (nbase + 48 + nl));
    c[0][0] = WMMA_F32(a0, b0, c[0][0]); c[1][0] = WMMA_F32(a1, b0, c[1][0]);
    c[0][1] = WMMA_F32(a0, b1, c[0][1]); c[1][1] = WMMA_F32(a1, b1, c[1][1]);
    c[0][2] = WMMA_F32(a0, b2, c[0][2]); c[1][2] = WMMA_F32(a1, b2, c[1][2]);
    c[0][3] = WMMA_F32(a0, b3, c[0][3]); c[1][3] = WMMA_F32(a1, b3, c[1][3]);
  }
  __syncthreads();  // all waves done reading As
#pragma unroll
  for (int s = 0; s < 2; ++s)
#pragma unroll
    for (int t = 0; t < 4; ++t)
#pragma unroll
      for (int j = 0; j < 8; ++j)
        AsY1[(s * 16 + mhi + j) * 256 + nbase + t * 16 + nl] = fmaxf(c[s][t][j], 0.f);
  __syncthreads();

  // GEMM2: (32x256)@Wc2(256x128) + bc2 -> relu -> Y2
  int nb2 = wave * 32;
  v8f d[2][2];
#pragma unroll
  for (int t = 0; t < 2; ++t) {
    float bb = bc2[nb2 + t * 16 + nl];
#pragma unroll
    for (int j = 0; j < 8; ++j) { d[0][t][j] = bb; d[1][t][j] = bb; }
  }
  for (int k = 0; k < 256; k += 4) {
    v2f a0 = ld2(AsY1 + nl * 256 + k + koff);
    v2f a1 = ld2(AsY1 + (nl + 16) * 256 + k + koff);
    const float* wt = Wc2t + (long)((k + koff) >> 1) * 256;
    v2f b0 = ld2(wt + 2 * (nb2 + nl));
    v2f b1 = ld2(wt + 2 * (nb2 + 16 + nl));
    d[0][0] = WMMA_F32(a0, b0, d[0][0]); d[1][0] = WMMA_F32(a1, b0, d[1][0]);
    d[0][1] = WMMA_F32(a0, b1, d[0][1]); d[1][1] = WMMA_F32(a1, b1, d[1][1]);
  }
#pragma unroll
  for (int s = 0; s < 2; ++s)
#pragma unroll
    for (int t = 0; t < 2; ++t)
#pragma unroll
      for (int j = 0; j < 8; ++j)
        Y2[(s * 16 + mhi + j) * 128 + nb2 + t * 16 + nl] = fmaxf(d[s][t][j], 0.f);
  __syncthreads();

  // final: y = Y2 @ Wc3 + bc3  (32 rows x 128 dot)
  if (threadIdx.x < 32) {
    float s = bc3[0];
    const float* yr = Y2 + threadIdx.x * 128;
    for (int k = 0; k < 128; ++k) s += yr[k] * Wc3[k];
    yscal[rbase + threadIdx.x] = s;
  }
}

// ------------------------------------------- masked mean over ctx -> out (B,1)
__global__ void reduce_kernel(const float* __restrict__ yscal, const int* __restrict__ indices,
                              float* __restrict__ out) {
  __shared__ float ssum[64];
  __shared__ float scnt[64];
  int b = blockIdx.x;
  int t = threadIdx.x;
  float s = 0.f, c = 0.f;
  if (t < CTXN) {
    int node = indices[b * IDXS + 3 + t];
    float m = (node > 0) ? 1.f : 0.f;
    s = yscal[b * CTXN + t] * m;
    c = m;
  }
  ssum[t] = s; scnt[t] = c;
  __syncthreads();
  for (int off = 32; off > 0; off >>= 1) {
    if (t < off) { ssum[t] += ssum[t + off]; scnt[t] += scnt[t + off]; }
    __syncthreads();
  }
  if (t == 0) out[b] = ssum[0] / scnt[0];
}

// ---------------------------------------------------------------------------
extern "C" void kernel_launch(void* const* d_in, const int* in_sizes, int n_in,
                              void* d_out, int out_size, void* d_ws, size_t ws_size,
                              hipStream_t stream) {
  const float* ndata  = (const float*)d_in[0];
  const int*   src    = (const int*)d_in[1];
  const int*   dst    = (const int*)d_in[2];
  const float* w      = (const float*)d_in[3];
  const float* eps    = (const float*)d_in[4];
  const float* W1     = (const float*)d_in[5];
  const float* b1     = (const float*)d_in[6];
  const float* W2     = (const float*)d_in[7];
  const float* b2     = (const float*)d_in[8];
  const float* Wc1    = (const float*)d_in[9];
  const float* bc1    = (const float*)d_in[10];
  const float* Wc2    = (const float*)d_in[11];
  const float* bc2    = (const float*)d_in[12];
  const float* Wc3    = (const float*)d_in[13];
  const float* bc3    = (const float*)d_in[14];
  const int*   indices = (const int*)d_in[15];
  float* out = (float*)d_out;

  // workspace (floats): deg | msg | xbuf | pair | yscal | W1t | W2t | Wc1t | Wc2t
  float* ws    = (float*)d_ws;
  float* deg   = ws;
  float* msg   = deg + 50048;
  float* xbuf  = msg + (long)NROWP * CH;
  float* pairb = xbuf + (long)NROWP * CH;
  float* yscal = pairb + (long)BATCH * CH;
  float* W1t   = yscal + (long)BATCH * CTXN;
  float* W2t   = W1t + 256 * 256;
  float* Wc1t  = W2t + 256 * 256;
  float* Wc2t  = Wc1t + 768 * 256;

  const long NM = (long)NROWP * CH;

  // ---- interleave weights (paired-row layout for b64 B-fragment loads)
  interleave_kernel<<<(128 * 256 + 255) / 256, 256, 0, stream>>>(W1, W1t, 128 * 256, 256);
  interleave_kernel<<<(128 * 256 + 255) / 256, 256, 0, stream>>>(W2, W2t, 128 * 256, 256);
  interleave_kernel<<<(384 * 256 + 255) / 256, 256, 0, stream>>>(Wc1, Wc1t, 384 * 256, 256);
  interleave_kernel<<<(128 * 128 + 255) / 256, 256, 0, stream>>>(Wc2, Wc2t, 128 * 128, 128);

  // ---- GIN layer 1
  zero_kernel<<<(unsigned)((50048 + NM + 255) / 256), 256, 0, stream>>>(deg, 50048 + NM);
  deg_kernel<<<N_EDGES / 256, 256, 0, stream>>>(dst, deg);
  scatter_kernel<<<(unsigned)(((long)N_EDGES * 64) / 256), 256, 0, stream>>>(ndata, src, dst, w, msg);
  h_kernel<<<(unsigned)(((long)N_NODES * 64 + 255) / 256), 256, 0, stream>>>(ndata, deg, eps, 0, msg);
  gemm256_kernel<<<NROWP / 32, 128, 0, stream>>>(msg, W1t, b1, xbuf, /*relu=*/1, /*zero0=*/0);

  // ---- GIN layer 2 (xbuf = relu(x1); x2 overwrites xbuf)
  zero_kernel<<<(unsigned)((NM + 255) / 256), 256, 0, stream>>>(msg, NM);
  scatter_kernel<<<(unsigned)(((long)N_EDGES * 64) / 256), 256, 0, stream>>>(xbuf, src, dst, w, msg);
  h_kernel<<<(unsigned)(((long)N_NODES * 64 + 255) / 256), 256, 0, stream>>>(xbuf, deg, eps, 1, msg);
  gemm256_kernel<<<NROWP / 32, 128, 0, stream>>>(msg, W2t, b2, xbuf, /*relu=*/0, /*zero0=*/1);

  // ---- pair + fused context MLP + masked mean
  pair_kernel<<<BATCH / 16, 128, 0, stream>>>(xbuf, indices, Wc1t, pairb);
  ctx_kernel<<<(BATCH * CTXN) / 32, 128, 0, stream>>>(xbuf, indices, Wc1t + 2 * 256 * 256, bc1,
                                                      pairb, Wc2t, bc2, Wc3, bc3, yscal);
  reduce_kernel<<<BATCH, 64, 0, stream>>>(yscal, indices, out);
}